// MWT_CZ_652835029835
// MI455X (gfx1250) — compile-verified
//
#include <hip/hip_runtime.h>
#include <math.h>

// ---------------------------------------------------------------------------
// MWT_CZ forward for MI455X (gfx1250, wave32).
// Truncated-DFT spectral path (MODES=5 => only 10x10x5 mode corner needed),
// all channel contractions via V_WMMA_F32_16X16X4_F32 (fp32 tensor path).
// GEMM shapes are compile-time template params so the WMMA chains fully
// unroll (no movrel, no divergent guarded loads).
// ---------------------------------------------------------------------------

typedef __attribute__((ext_vector_type(2))) float v2f;
typedef __attribute__((ext_vector_type(8))) float v8f;

#define PI_F 3.14159265358979f
#define CDIV(a, b) (((a) + (b) - 1) / (b))

constexpr int MODES = 5;

// ---- workspace layout (float element offsets) -----------------------------
constexpr long SZ_XA  = 16L * 36 * 32 * 32 * 64;   // 37,748,736 (also 4 recon tmps)
constexpr long SZ_F   = 16L * 9 * 32 * 32 * 64;    //  9,437,184
constexpr long SZ_FT1 = 16L * 9 * 32 * 32 * 5 * 2; //  1,474,560
constexpr long SZ_FT2 = 16L * 9 * 32 * 10 * 5 * 2; //    460,800
constexpr long SZ_FT3 = 16L * 9 * 10 * 10 * 5 * 2; //    144,000
constexpr long SZ_UD  = SZ_F + SZ_F/4 + SZ_F/16 + SZ_F/64 + SZ_F/256 + SZ_F/1024;

constexpr long O_XA  = 0;
constexpr long O_X   = O_XA + SZ_XA;
constexpr long O_D   = O_X  + SZ_F;
constexpr long O_S   = O_D  + SZ_F;   // CAT (18ch) aliases O_D..O_S (exactly 2*SZ_F)
constexpr long O_R   = O_S  + SZ_F;
constexpr long O_FT1 = O_R  + SZ_F;
constexpr long O_FT6 = O_FT1 + SZ_FT1;
constexpr long O_FT2 = O_FT6 + SZ_FT1;
constexpr long O_FT5 = O_FT2 + SZ_FT2;
constexpr long O_FT3 = O_FT5 + SZ_FT2;
constexpr long O_FT4 = O_FT3 + SZ_FT3;
constexpr long O_UD  = O_FT4 + SZ_FT3;
constexpr long O_US  = O_UD + SZ_UD;

// ---------------------------------------------------------------------------
// WMMA channel-mix GEMM:  out[b,o,s] = sum_k W(o,k) * f(in[b,k,s]) (+bias)(+acc)
// Per wave: 16 output channels x 16 positions, K padded to 4*KSTEPS.
// fp32 WMMA 16x16x4: A lanes0-15 -> M=lane, K={k0,k0+1}; lanes16-31 K={k0+2,k0+3}.
// B: VGPR v, lanes0-15 -> K=v, lanes16-31 -> K=v+2, col = lane&15.
// D: VGPR j, lanes0-15 -> row j, lanes16-31 -> row j+8, col = lane&15.
// Out-of-range K lanes use clamped (always valid) addresses + value select,
// so the inner loop has no exec-mask branches.
// ---------------------------------------------------------------------------
template<int CHI, int CHO, int W_OI>
__global__ void k_chan_mix(const float* __restrict__ in, const float* __restrict__ w,
                           const float* __restrict__ bias, float* __restrict__ out,
                           long S, int do_relu, int accum)
{
    constexpr int KSTEPS = (CHI + 3) / 4;
    int  lane = threadIdx.x & 31;
    long wave = (long)blockIdx.x * (blockDim.x >> 5) + (threadIdx.x >> 5);
    long n0   = wave * 16;
    if (n0 >= 16L * S) return;                 // wave-uniform: EXEC all-1s at WMMA
    int  b  = (int)(n0 / S);
    long s0 = n0 % S;

    const float* inb = in + (long)b * CHI * S + s0;
    __builtin_prefetch(inb);                   // global_prefetch_b8

    int col = lane & 15;
    int m   = lane & 15;
    int kh  = (lane >> 4) * 2;
    int kb  = (lane < 16) ? 0 : 2;
    bool mok = (m < CHO);
    int  mm  = mok ? m : 0;

    v2f a[KSTEPS];
#pragma unroll
    for (int st = 0; st < KSTEPS; ++st) {
        int k0 = st * 4 + kh;
        int k1 = k0 + 1;
        int c0 = (k0 < CHI) ? k0 : 0;
        int c1 = (k1 < CHI) ? k1 : 0;
        float a0 = W_OI ? w[mm * CHI + c0] : w[c0 * CHO + mm];
        float a1 = W_OI ? w[mm * CHI + c1] : w[c1 * CHO + mm];
        a[st].x = (mok && k0 < CHI) ? a0 : 0.f;
        a[st].y = (mok && k1 < CHI) ? a1 : 0.f;
    }

    v8f c = {};
#pragma unroll
    for (int st = 0; st < KSTEPS; ++st) {
        int k0 = st * 4 + kb;
        int k1 = k0 + 1;
        int c0 = (k0 < CHI) ? k0 : 0;          // clamp: load always valid
        int c1 = (k1 < CHI) ? k1 : 0;
        float b0 = inb[(long)c0 * S + col];
        float b1 = inb[(long)c1 * S + col];
        if (do_relu) { b0 = fmaxf(b0, 0.f); b1 = fmaxf(b1, 0.f); }
        b0 = (k0 < CHI) ? b0 : 0.f;            // zero the padded K lanes
        b1 = (k1 < CHI) ? b1 : 0.f;
        v2f bv; bv.x = b0; bv.y = b1;
        c = __builtin_amdgcn_wmma_f32_16x16x4_f32(false, a[st], false, bv,
                                                  (short)0, c, false, false);
    }

    float* ob  = out + (long)b * CHO * S + s0 + col;
    int    ob0 = (lane < 16) ? 0 : 8;
#pragma unroll
    for (int j = 0; j < 8; ++j) {
        int o = ob0 + j;
        if (o < CHO) {
            float v = c[j];
            if (bias)  v += bias[o];
            if (accum) v += ob[(long)o * S];
            ob[(long)o * S] = v;
        }
    }
}

// ---------------------------------------------------------------------------
// quadrant gather: x[b,9,n,n,T] -> xa[b,36,h,h,T]  (quadrant order matches
// reference concat: (e,e),(e,o),(o,e),(o,o))
// ---------------------------------------------------------------------------
__global__ void k_quad_gather(const float* __restrict__ src, float* __restrict__ dst,
                              int h, long total)
{
    long i = (long)blockIdx.x * blockDim.x + threadIdx.x;
    if (i >= total) return;
    int t = (int)(i & 63);
    long r = i >> 6;
    int y = (int)(r % h); r /= h;
    int x = (int)(r % h); r /= h;
    int ch = (int)(r % 36);
    long b = r / 36;
    int q = ch / 9, ic = ch % 9;
    int n = 2 * h;
    int sx = 2 * x + (q >> 1), sy = 2 * y + (q & 1);
    dst[i] = src[(((b * 9 + ic) * n + sx) * (long)n + sy) * 64 + t];
}

// ---------------------------------------------------------------------------
// truncated DFT stages (ortho norm folded per axis)
// ---------------------------------------------------------------------------
__global__ void k_dft_t_fwd(const float* __restrict__ in, float2* __restrict__ out,
                            long rows, float scale)
{
    long i = (long)blockIdx.x * blockDim.x + threadIdx.x;
    if (i >= rows * MODES) return;
    int  kz  = (int)(i % MODES);
    long row = i / MODES;
    const float* p = in + row * 64;
    float w = -2.f * PI_F * kz / 64.f;
    float re = 0.f, im = 0.f;
#pragma unroll 4
    for (int t = 0; t < 64; ++t) {
        float s, c; __sincosf(w * t, &s, &c);
        re += p[t] * c; im += p[t] * s;
    }
    out[i] = make_float2(re * scale, im * scale);
}

__device__ __forceinline__ int mode_freq(int m, int l, int n, int nm)
{
    return (nm == n) ? m : (m < l ? m : n - 2 * l + m);
}

__global__ void k_dft_c_fwd(const float2* __restrict__ in, float2* __restrict__ out,
                            long outer, int n, int nm, int l, int inner, float scale)
{
    long i = (long)blockIdx.x * blockDim.x + threadIdx.x;
    if (i >= outer * nm * inner) return;
    int  ii = (int)(i % inner);
    int  m  = (int)((i / inner) % nm);
    long o  = i / ((long)inner * nm);
    int  f  = mode_freq(m, l, n, nm);
    const float2* src = in + o * (long)n * inner + ii;
    float w = -2.f * PI_F * f / n;
    float re = 0.f, im = 0.f;
    for (int j = 0; j < n; ++j) {
        float s, c; __sincosf(w * j, &s, &c);
        float2 v = src[(long)j * inner];
        re += v.x * c - v.y * s;
        im += v.x * s + v.y * c;
    }
    out[i] = make_float2(re * scale, im * scale);
}

__global__ void k_dft_c_inv(const float2* __restrict__ in, float2* __restrict__ out,
                            long outer, int n, int nm, int l, int inner, float scale)
{
    long i = (long)blockIdx.x * blockDim.x + threadIdx.x;
    if (i >= outer * (long)n * inner) return;
    int  ii = (int)(i % inner);
    int  j  = (int)((i / inner) % n);
    long o  = i / ((long)inner * n);
    const float2* src = in + o * (long)nm * inner + ii;
    float re = 0.f, im = 0.f;
    for (int m = 0; m < nm; ++m) {
        int f = mode_freq(m, l, n, nm);
        float th = 2.f * PI_F * f * j / n;
        float s, c; __sincosf(th, &s, &c);
        float2 v = src[(long)m * inner];
        re += v.x * c - v.y * s;
        im += v.x * s + v.y * c;
    }
    out[i] = make_float2(re * scale, im * scale);
}

__global__ void k_dft_t_inv(const float2* __restrict__ in, float* __restrict__ out,
                            long rows, float scale)
{
    long i = (long)blockIdx.x * blockDim.x + threadIdx.x;
    if (i >= rows * 64) return;
    int  t   = (int)(i & 63);
    long row = i >> 6;
    const float2* p = in + row * MODES;
    float acc = p[0].x;
#pragma unroll
    for (int k = 1; k < MODES; ++k) {
        float th = 2.f * PI_F * k * t / 64.f;
        float s, c; __sincosf(th, &s, &c);
        acc += 2.f * (p[k].x * c - p[k].y * s);
    }
    out[i] = acc * scale;
}

// ---------------------------------------------------------------------------
// mode mix: out[b,o,mode] = sum_i in[b,i,mode] * W_q[i,o,kx',ky',kz] (complex)
// Real K=18 GEMM (Re/Im concat), M = batch = 16, N = out-ch (9, padded 16).
// Quadrant priority q4>q3>q2>q1 replicates sequential overwrite semantics.
// All lane-dependent bounds use clamp+select (no divergent loads).
// ---------------------------------------------------------------------------
__device__ __forceinline__ float mix_a(const float2* __restrict__ in, int b, int k,
                                       int mi, int nmodes)
{
    int kk = (k < 18) ? k : 0;
    int ch = (kk < 9) ? kk : kk - 9;
    float2 v = in[((long)b * 9 + ch) * nmodes + mi];
    float r = (kk < 9) ? v.x : v.y;
    return (k < 18) ? r : 0.f;
}

__global__ void k_mode_mix(const float2* __restrict__ in, float2* __restrict__ out,
                           const float* __restrict__ w1, const float* __restrict__ w2,
                           const float* __restrict__ w3, const float* __restrict__ w4,
                           int nx, int ny, int l1, int l2, int nmx, int nmy)
{
    int lane   = threadIdx.x & 31;
    int nmodes = nmx * nmy * MODES;
    int mi = blockIdx.x * (blockDim.x >> 5) + (threadIdx.x >> 5);
    if (mi >= nmodes) return;                  // wave-uniform

    int kz  = mi % MODES;
    int kyI = (mi / MODES) % nmy;
    int kxI = mi / (MODES * nmy);
    int kx  = mode_freq(kxI, l1, nx, nmx);
    int ky  = mode_freq(kyI, l2, ny, nmy);
    bool hx = kx >= nx - l1, hy = ky >= ny - l2;
    const float* w; int kxw, kyw;
    if (hx && hy) { w = w4; kxw = kx - (nx - l1); kyw = ky - (ny - l2); }
    else if (hy)  { w = w3; kxw = kx;             kyw = ky - (ny - l2); }
    else if (hx)  { w = w2; kxw = kx - (nx - l1); kyw = ky; }
    else          { w = w1; kxw = kx;             kyw = ky; }
    long wmode = (((long)kxw * MODES + kyw) * MODES + kz) * 2;   // (.,.,5,5,5,2)

    int m  = lane & 15;              // batch row for A
    int kh = (lane >> 4) * 2;
    int o  = lane & 15;              // out-channel col for B/D
    int wo = (o < 9) ? o : 8;        // clamped weight column
    int kb = (lane < 16) ? 0 : 2;

    v8f cre = {}, cim = {};
#pragma unroll
    for (int st = 0; st < 5; ++st) {           // K = 18 padded to 20
        int ka = st * 4 + kh;
        v2f a;
        a.x = mix_a(in, m, ka,     mi, nmodes);
        a.y = mix_a(in, m, ka + 1, mi, nmodes);
        int k0 = st * 4 + kb;
        float brv[2], biv[2];
#pragma unroll
        for (int u = 0; u < 2; ++u) {
            int k  = k0 + u;
            int kk = (k < 18) ? k : 0;
            int ki = (kk < 9) ? kk : kk - 9;
            const float* wp = w + ((long)ki * 9 + wo) * 250 + wmode;
            float wr = wp[0], wi_ = wp[1];
            float r_  = (kk < 9) ? wr  : -wi_;
            float im_ = (kk < 9) ? wi_ :  wr;
            bool ok = (o < 9) && (k < 18);
            brv[u] = ok ? r_  : 0.f;
            biv[u] = ok ? im_ : 0.f;
        }
        v2f br; br.x = brv[0]; br.y = brv[1];
        v2f bi; bi.x = biv[0]; bi.y = biv[1];
        cre = __builtin_amdgcn_wmma_f32_16x16x4_f32(false, a, false, br, (short)0, cre, false, false);
        cim = __builtin_amdgcn_wmma_f32_16x16x4_f32(false, a, false, bi, (short)0, cim, false, false);
    }

    if (o < 9) {
        int bb0 = (lane < 16) ? 0 : 8;
#pragma unroll
        for (int j = 0; j < 8; ++j) {
            int b = bb0 + j;
            out[((long)b * 9 + o) * nmodes + mi] = make_float2(cre[j], cim[j]);
        }
    }
}

// ---------------------------------------------------------------------------
// recon helpers
// ---------------------------------------------------------------------------
__global__ void k_cat(const float* __restrict__ x, const float* __restrict__ us,
                      const float* __restrict__ ud, float* __restrict__ dst,
                      long S, long total)
{
    long i = (long)blockIdx.x * blockDim.x + threadIdx.x;
    if (i >= total) return;
    long s = i % S;
    long r = i / S;
    int  ch = (int)(r % 18);
    long b  = r / 18;
    if (ch < 9) dst[i] = x[(b * 9 + ch) * S + s] + us[(b * 9 + ch) * S + s];
    else        dst[i] = ud[(b * 9 + (ch - 9)) * S + s];
}

__global__ void k_scatter(const float* __restrict__ ee, const float* __restrict__ eo,
                          const float* __restrict__ oe, const float* __restrict__ oo,
                          float* __restrict__ out, int hn, long total)
{
    long i = (long)blockIdx.x * blockDim.x + threadIdx.x;
    if (i >= total) return;
    int t = (int)(i & 63);
    long r = i >> 6;
    int y = (int)(r % hn); r /= hn;
    int x = (int)(r % hn); r /= hn;      // r = b*9+ch
    int n2 = hn * 2;
    long base = ((r * n2 + 2 * x) * (long)n2 + 2 * y) * 64 + t;
    out[base]                       = ee[i];
    out[base + 64]                  = eo[i];
    out[base + (long)n2 * 64]       = oe[i];
    out[base + (long)n2 * 64 + 64]  = oo[i];
}

// ---------------------------------------------------------------------------
// host side
// ---------------------------------------------------------------------------
static inline void launch_mix(const float* in, const float* w, const float* bias,
                              float* out, long S, int CHI, int w_oi,
                              int relu, int accum, hipStream_t st)
{
    unsigned blocks = (unsigned)CDIV(S, 4);    // 1 wave = 16 positions, 4 waves/block
    if (CHI == 36)
        k_chan_mix<36, 9, 0><<<dim3(blocks), 128, 0, st>>>(in, w, bias, out, S, relu, accum);
    else if (CHI == 18)
        k_chan_mix<18, 9, 0><<<dim3(blocks), 128, 0, st>>>(in, w, bias, out, S, relu, accum);
    else if (w_oi)
        k_chan_mix<9, 9, 1><<<dim3(blocks), 128, 0, st>>>(in, w, bias, out, S, relu, accum);
    else
        k_chan_mix<9, 9, 0><<<dim3(blocks), 128, 0, st>>>(in, w, bias, out, S, relu, accum);
}

static void run_spectral(const float* src, const float* const wq[4],
                         const float* lo_w, const float* lo_b,
                         float* dst, int accum, int n, float* ws, hipStream_t st)
{
    int l = n / 2 + 1; if (l > MODES) l = MODES;
    int nm = (n < 2 * l) ? n : 2 * l;
    float sc = 1.f / sqrtf((float)n);
    float2* FT1 = (float2*)(ws + O_FT1);
    float2* FT2 = (float2*)(ws + O_FT2);
    float2* FT3 = (float2*)(ws + O_FT3);
    float2* FT4 = (float2*)(ws + O_FT4);
    float2* FT5 = (float2*)(ws + O_FT5);
    float2* FT6 = (float2*)(ws + O_FT6);
    long rows = 16L * 9 * n * n;

    k_dft_t_fwd<<<dim3((unsigned)CDIV(rows * MODES, 256)), 256, 0, st>>>(src, FT1, rows, 0.125f);
    long t2 = 16L * 9 * n * nm * 5;
    k_dft_c_fwd<<<dim3((unsigned)CDIV(t2, 256)), 256, 0, st>>>(FT1, FT2, 16L * 9 * n, n, nm, l, 5, sc);
    long t3 = 16L * 9 * (long)nm * nm * 5;
    k_dft_c_fwd<<<dim3((unsigned)CDIV(t3, 256)), 256, 0, st>>>(FT2, FT3, 16L * 9, n, nm, l, nm * 5, sc);
    int nmodes = nm * nm * MODES;
    k_mode_mix<<<dim3((unsigned)CDIV(nmodes, 4)), 128, 0, st>>>(FT3, FT4, wq[0], wq[1], wq[2], wq[3],
                                                               n, n, l, l, nm, nm);
    long t5 = 16L * 9 * n * (long)nm * 5;
    k_dft_c_inv<<<dim3((unsigned)CDIV(t5, 256)), 256, 0, st>>>(FT4, FT5, 16L * 9, n, nm, l, nm * 5, sc);
    long t6 = 16L * 9 * n * (long)n * 5;
    k_dft_c_inv<<<dim3((unsigned)CDIV(t6, 256)), 256, 0, st>>>(FT5, FT6, 16L * 9 * n, n, nm, l, 5, sc);
    k_dft_t_inv<<<dim3((unsigned)CDIV(rows * 64, 256)), 256, 0, st>>>(FT6, ws + O_R, rows, 0.125f);
    launch_mix(ws + O_R, lo_w, lo_b, dst, (long)n * n * 64, 9, 1, 1, accum, st);
}

extern "C" void kernel_launch(void* const* d_in, const int* in_sizes, int n_in,
                              void* d_out, int out_size, void* d_ws, size_t ws_size,
                              hipStream_t stream)
{
    (void)in_sizes; (void)n_in; (void)out_size; (void)ws_size;
    const float* x_in = (const float*)d_in[0];
    const float* sw[3][4]; const float* slo[3]; const float* slob[3];
    int idx = 1;
    for (int s = 0; s < 3; ++s) {
        for (int j = 0; j < 4; ++j) sw[s][j] = (const float*)d_in[idx++];
        slo[s]  = (const float*)d_in[idx++];
        slob[s] = (const float*)d_in[idx++];
    }
    const float* t0_w = (const float*)d_in[idx++];
    const float* t0_b = (const float*)d_in[idx++];
    const float* ec_s = (const float*)d_in[idx++];
    const float* ec_d = (const float*)d_in[idx++];
    const float* rc[4];
    for (int q = 0; q < 4; ++q) rc[q] = (const float*)d_in[idx++];

    float* ws = (float*)d_ws;
    long udoff[6]; { long a = 0; for (int l = 0; l < 6; ++l) { udoff[l] = a; a += SZ_F >> (2 * l); } }

    // ---------------- forward (decomposition) ----------------
    for (int lev = 0; lev < 6; ++lev) {
        int n = 64 >> lev, h = n >> 1;
        long Sh = (long)h * h * 64;
        long totXA = 16L * 36 * Sh;
        const float* xsrc = (lev == 0) ? x_in : (ws + O_X);
        k_quad_gather<<<dim3((unsigned)CDIV(totXA, 256)), 256, 0, stream>>>(xsrc, ws + O_XA, h, totXA);
        launch_mix(ws + O_XA, ec_d, nullptr, ws + O_D, Sh, 36, 0, 0, 0, stream);
        launch_mix(ws + O_XA, ec_s, nullptr, ws + O_S, Sh, 36, 0, 0, 0, stream);
        run_spectral(ws + O_D, sw[0], slo[0], slob[0], ws + O_UD + udoff[lev], 0, h, ws, stream);
        run_spectral(ws + O_S, sw[1], slo[1], slob[1], ws + O_UD + udoff[lev], 1, h, ws, stream);
        run_spectral(ws + O_D, sw[2], slo[2], slob[2], ws + O_US + udoff[lev], 0, h, ws, stream);
        hipMemcpyAsync(ws + O_X, ws + O_S, sizeof(float) * 16L * 9 * Sh,
                       hipMemcpyDeviceToDevice, stream);
    }

    // ---------------- coarsest 1x1 conv (in place) ----------------
    launch_mix(ws + O_X, t0_w, t0_b, ws + O_X, 64, 9, 1, 0, 0, stream);

    // ---------------- reconstruction ----------------
    float* CAT = ws + O_D;               // aliases D+S (forward-only buffers)
    for (int i = 5; i >= 0; --i) {
        int hn = 32 >> i;
        long S = (long)hn * hn * 64;
        long totC = 16L * 18 * S;
        k_cat<<<dim3((unsigned)CDIV(totC, 256)), 256, 0, stream>>>(ws + O_X, ws + O_US + udoff[i],
                                                                   ws + O_UD + udoff[i], CAT, S, totC);
        float* tmp[4];
        for (int q = 0; q < 4; ++q) {
            tmp[q] = ws + O_XA + (long)q * SZ_F;
            launch_mix(CAT, rc[q], nullptr, tmp[q], S, 18, 0, 0, 0, stream);
        }
        float* dstx = (i == 0) ? (float*)d_out : (ws + O_X);
        long totS = 16L * 9 * S;
        k_scatter<<<dim3((unsigned)CDIV(totS, 256)), 256, 0, stream>>>(tmp[0], tmp[1], tmp[2], tmp[3],
                                                                       dstx, hn, totS);
    }
}